// Loss_fun_11338713661811
// MI455X (gfx1250) — compile-verified
//
#include <hip/hip_runtime.h>
#include <cstdint>

#define NCOL 128
#define BLOCK_THREADS 256
#define WAVES_PER_BLOCK (BLOCK_THREADS / 32)
#define GRID_BLOCKS 4096

// gfx1250 async global->LDS path (guarded so the file always compiles; the
// fallback uses plain global_load_b128).
#if defined(__HIP_DEVICE_COMPILE__) &&                                         \
    __has_builtin(__builtin_amdgcn_global_load_async_to_lds_b128) &&           \
    __has_builtin(__builtin_amdgcn_s_wait_asynccnt)
#define USE_ASYNC_LDS 1
#else
#define USE_ASYNC_LDS 0
#endif

#if USE_ASYNC_LDS
// Builtin params: pointer to 4xi32 vector in AS1 (global) and AS3 (LDS).
typedef int v4i __attribute__((ext_vector_type(4)));
typedef __attribute__((address_space(1))) v4i av1_v4i;
typedef __attribute__((address_space(3))) v4i av3_v4i;

__device__ __forceinline__ av1_v4i* as_global(const void* p) {
  return (av1_v4i*)(uintptr_t)p;  // flat addr == global addr
}
__device__ __forceinline__ av3_v4i* as_lds(const void* p) {
  return (av3_v4i*)(uint32_t)(uintptr_t)p;  // ISA 10.2: LDS address is the low 32 bits
}
#endif

__device__ __forceinline__ float wave_max(float v) {
#pragma unroll
  for (int m = 16; m > 0; m >>= 1) v = fmaxf(v, __shfl_xor(v, m, 32));
  return v;
}

__device__ __forceinline__ float wave_sum(float v) {
#pragma unroll
  for (int m = 16; m > 0; m >>= 1) v += __shfl_xor(v, m, 32);
  return v;
}

__device__ __forceinline__ unsigned int ballot32(bool p) {
#if defined(__HIP_DEVICE_COMPILE__) && __has_builtin(__builtin_amdgcn_ballot_w32)
  return __builtin_amdgcn_ballot_w32(p);
#else
  return (unsigned int)__ballot(p);
#endif
}

// One wave32 per row (128 cols = 32 lanes x float4). t0 is wave-uniform, so
// the is_null branch costs no divergence and roughly halves VALU work.
__device__ __forceinline__ float row_loss(const float4 xv, const int4 tv,
                                          float x0, int t0) {
  const float NEG_BIG = -3.402823466e38f;  // finfo(float32).min
  float xa[4] = {xv.x, xv.y, xv.z, xv.w};
  int ta[4] = {tv.x, tv.y, tv.z, tv.w};

  if (t0 != 0) {
    // is_null: loss_simple = logsumexp(all) - x0
    float m = NEG_BIG;
#pragma unroll
    for (int e = 0; e < 4; ++e) m = fmaxf(m, xa[e]);
    m = wave_max(m);
    float s = 0.f;
#pragma unroll
    for (int e = 0; e < 4; ++e) s += __expf(xa[e] - m);
    s = wave_sum(s);
    return m + __logf(s) - x0;
  }

  // full path (t0 == 0): mask_pos = t | {col0}, mask_neg = ~t (col0 in ~t).
  float mp = NEG_BIG, mn = NEG_BIG;
#pragma unroll
  for (int e = 0; e < 4; ++e) {
    const bool tb = ta[e] != 0;
    if (tb) mp = fmaxf(mp, xa[e]);
    else    mn = fmaxf(mn, xa[e]);
  }
  mp = fmaxf(wave_max(mp), x0);  // forced col0 membership in mask_pos
  mn = fmaxf(wave_max(mn), x0);  // no-op here (t0==0) but matches semantics

  float sp = 0.f, sn = 0.f, sx = 0.f;
  unsigned int np_i = 0;
#pragma unroll
  for (int e = 0; e < 4; ++e) {
    const bool tb = ta[e] != 0;
    const float ex = __expf(xa[e] - (tb ? mp : mn));  // 1 exp per element
    sp += tb ? ex : 0.0f;
    sn += tb ? 0.0f : ex;
    sx += tb ? xa[e] : 0.0f;            // pos_no0 sum (col0 has tb==false)
    np_i += __popc(ballot32(tb));       // n_pos via ballot+popc (SGPR side)
  }
  sp = wave_sum(sp) + __expf(x0 - mp);  // col0 term of mask_pos
  sn = wave_sum(sn);
  sx = wave_sum(sx);
  const float np = (float)np_i;         // uniform across wave

  const float lse_pos = mp + __logf(sp);
  const float lse_neg = mn + __logf(sn);
  const float loss_positive = (np * lse_pos - sx) / fmaxf(np, 1.0f);
  return loss_positive + lse_neg - x0;
}

__global__ __launch_bounds__(BLOCK_THREADS) void bce_lse_loss_kernel(
    const float* __restrict__ logit, const int* __restrict__ target,
    float* __restrict__ partial, int nrows) {
  const int lane = threadIdx.x & 31;
  const int wave = threadIdx.x >> 5;
  const int gw = blockIdx.x * WAVES_PER_BLOCK + wave;
  const int stride = gridDim.x * WAVES_PER_BLOCK;
  float acc = 0.0f;

#if USE_ASYNC_LDS
  // Per-wave private double buffers: no barriers, only ASYNCcnt waits.
  __shared__ float s_log[2][WAVES_PER_BLOCK][NCOL];  // 8 KB
  __shared__ int s_tgt[2][WAVES_PER_BLOCK][NCOL];    // 8 KB

  long long row = gw;
  if (row < nrows) {  // prologue prefetch into buffer 0 (2 async ops)
    __builtin_amdgcn_global_load_async_to_lds_b128(
        as_global(logit + row * NCOL + lane * 4),
        as_lds(&s_log[0][wave][lane * 4]), 0, 0);
    __builtin_amdgcn_global_load_async_to_lds_b128(
        as_global(target + row * NCOL + lane * 4),
        as_lds(&s_tgt[0][wave][lane * 4]), 0, 0);
  }
  int buf = 0;
  for (; row < nrows; row += stride) {
    const long long nxt = row + stride;
    const bool pf = nxt < nrows;  // wave-uniform
    if (pf) {  // prefetch next row into the other buffer, then wait for current
      __builtin_amdgcn_global_load_async_to_lds_b128(
          as_global(logit + nxt * NCOL + lane * 4),
          as_lds(&s_log[buf ^ 1][wave][lane * 4]), 0, 0);
      __builtin_amdgcn_global_load_async_to_lds_b128(
          as_global(target + nxt * NCOL + lane * 4),
          as_lds(&s_tgt[buf ^ 1][wave][lane * 4]), 0, 0);
      __builtin_amdgcn_s_wait_asynccnt(2);  // async ops complete in order
    } else {
      __builtin_amdgcn_s_wait_asynccnt(0);
    }
    asm volatile("" ::: "memory");  // keep LDS reads below the wait

    const float4 xv =
        *reinterpret_cast<const float4*>(&s_log[buf][wave][lane * 4]);
    const int4 tv = *reinterpret_cast<const int4*>(&s_tgt[buf][wave][lane * 4]);
    const float x0 = s_log[buf][wave][0];  // free broadcast from LDS
    const int t0 = s_tgt[buf][wave][0];
    acc += row_loss(xv, tv, x0, t0);
    buf ^= 1;
  }
#else
  for (long long row = gw; row < nrows; row += stride) {
    const float4 xv =
        *reinterpret_cast<const float4*>(logit + row * NCOL + lane * 4);
    const int4 tv =
        *reinterpret_cast<const int4*>(target + row * NCOL + lane * 4);
    const float x0 = __shfl(xv.x, 0, 32);
    const int t0 = __shfl(tv.x, 0, 32);
    acc += row_loss(xv, tv, x0, t0);
  }
#endif

  // Deterministic in-block combine: ordered 8-way sum by thread 0.
  __shared__ float wsum[WAVES_PER_BLOCK];
  if (lane == 0) wsum[wave] = acc;
  __syncthreads();
  if (threadIdx.x == 0) {
    float s = 0.f;
#pragma unroll
    for (int i = 0; i < WAVES_PER_BLOCK; ++i) s += wsum[i];
    partial[blockIdx.x] = s;
  }
}

// Deterministic final reduction: fixed strided sums + fixed tree pairing.
__global__ __launch_bounds__(256) void final_reduce_kernel(
    const float* __restrict__ partial, float* __restrict__ out, int n,
    float inv_b) {
  __shared__ float sm[256];
  float s = 0.f;
  for (int i = threadIdx.x; i < n; i += 256) s += partial[i];
  sm[threadIdx.x] = s;
  __syncthreads();
  for (int off = 128; off > 0; off >>= 1) {
    if ((int)threadIdx.x < off) sm[threadIdx.x] += sm[threadIdx.x + off];
    __syncthreads();
  }
  if (threadIdx.x == 0) out[0] = sm[0] * inv_b;
}

extern "C" void kernel_launch(void* const* d_in, const int* in_sizes, int n_in,
                              void* d_out, int out_size, void* d_ws,
                              size_t ws_size, hipStream_t stream) {
  (void)n_in;
  (void)out_size;
  const float* logit = (const float*)d_in[0];
  const int* target = (const int*)d_in[1];
  float* out = (float*)d_out;
  float* partial = (float*)d_ws;

  const int nrows = in_sizes[0] / NCOL;  // 524288

  int blocks = GRID_BLOCKS;
  if (ws_size < (size_t)blocks * sizeof(float)) {
    blocks = (int)(ws_size / sizeof(float));
    if (blocks < 1) blocks = 1;
  }

  bce_lse_loss_kernel<<<blocks, BLOCK_THREADS, 0, stream>>>(logit, target,
                                                            partial, nrows);
  final_reduce_kernel<<<1, 256, 0, stream>>>(partial, out, blocks,
                                             1.0f / (float)nrows);
}